// OptionRetriever_14903536517725
// MI455X (gfx1250) — compile-verified
//
#include <hip/hip_runtime.h>

// CDNA5 (gfx1250) ColBERT max-sim:  out[b,o,d] = sum_u max_v <hq[b,o,u,:], hd[b,o,d,v,:]>
// hq: [16,4,128,128] f32, hd: [16,4,16,256,128] f32, out: [16,4,16] f32 (1024 scalars)
//
// One block per (b,o,d). 8 waves; each wave owns 16 query rows with its A panel
// register-resident (WMMA f32 16x16x4 A layout). Doc rows stream through a
// ping-pong LDS buffer via async global->LDS loads (ASYNCcnt-tracked), padded
// stride 132 so B-operand ds_load_b64 reads are bank-conflict-free. The two
// 16-wide v-tiles per chunk run as independent interleaved WMMA chains.

typedef __attribute__((ext_vector_type(2))) float v2f;
typedef __attribute__((ext_vector_type(8))) float v8f;
typedef __attribute__((ext_vector_type(4))) int   v4i;

#define WAVES 8
#define ROWS 32                 // doc rows per chunk
#define NCHUNK 8                // 256 doc rows / 32
#define LDS_STRIDE 132          // 128 + 4 floats pad: rows 16B-aligned, conflict-free reads

#if __has_builtin(__builtin_amdgcn_global_load_async_to_lds_b128)
#define HAVE_ASYNC_LDS 1
typedef __attribute__((address_space(1))) v4i gv4i;   // global (__device__) int4
typedef __attribute__((address_space(3))) v4i lv4i;   // LDS (__shared__) int4
#else
#define HAVE_ASYNC_LDS 0
#endif

#if __has_builtin(__builtin_amdgcn_s_wait_asynccnt)
#define WAIT_ASYNC(n) __builtin_amdgcn_s_wait_asynccnt(n)
#else
#define WAIT_ASYNC(n) asm volatile("s_wait_asynccnt %0" ::"i"(n) : "memory")
#endif

__global__ __launch_bounds__(256)
void colbert_maxsim_kernel(const float* __restrict__ hq,
                           const float* __restrict__ hd,
                           float* __restrict__ out) {
    __shared__ float ldsD[2][ROWS * LDS_STRIDE];   // ping-pong doc chunks (2 x 16.9 KB)
    __shared__ float red[WAVES];

    const int tid  = threadIdx.x;
    const int lane = tid & 31;
    const int wave = tid >> 5;
    const int M    = lane & 15;   // row within 16x16 tile (A: M, B/C: N)
    const int kh   = lane >> 4;   // lane half selects the K pair (A/B operand layout)

    const int bid = blockIdx.x;          // bid = ((b*4 + o)*16 + d)
    const int bo  = bid >> 4;            // b*4 + o
    const long long qbase = (long long)bo  * (128LL * 128LL);
    const long long dbase = (long long)bid * (256LL * 128LL);

    const float4* src = (const float4*)(hd + dbase);   // 1024 float4 per 32-row chunk

    // ---- Stage one 32x128 f32 chunk (16 KB) into ldsD[buf]: 4 x b128 per thread ----
    auto stage = [&](int chunk, int buf) {
#pragma unroll
        for (int i = 0; i < 4; ++i) {
            int e  = i * 256 + tid;      // float4 index within chunk (coalesced)
            int r  = e >> 5;             // 32 float4 per doc row
            int c4 = e & 31;
#if HAVE_ASYNC_LDS
            __builtin_amdgcn_global_load_async_to_lds_b128(
                (gv4i*)(uintptr_t)(src + chunk * 1024 + e),
                (lv4i*)(uintptr_t)&ldsD[buf][r * LDS_STRIDE + c4 * 4],
                /*offset=*/0, /*cpol=*/0);
#else
            float4 v = src[chunk * 1024 + e];
            *(float4*)&ldsD[buf][r * LDS_STRIDE + c4 * 4] = v;
#endif
        }
    };

    // ---- Preload this wave's A panel (16 query rows x K=128) into 64 VGPRs ----
    // ISA 16x4 f32 A layout: lane holds M = lane%16; VGPR0 = K(2*half), VGPR1 = K(2*half+1).
    v2f a[32];
    {
        const float* qrow = hq + qbase + (long long)(wave * 16 + M) * 128 + kh * 2;
#pragma unroll
        for (int kk = 0; kk < 32; ++kk)
            a[kk] = *(const v2f*)(qrow + kk * 4);
    }

    // Running max over all v, kept in C-matrix register layout.
    v8f rowmax;
#pragma unroll
    for (int r = 0; r < 8; ++r) rowmax[r] = -3.402823466e38f;

    stage(0, 0);                          // pipeline prologue

    for (int chunk = 0; chunk < NCHUNK; ++chunk) {
        const int cur = chunk & 1;

        if (chunk + 1 < NCHUNK) {
            stage(chunk + 1, cur ^ 1);    // issue next chunk before computing this one
            if (chunk + 2 < NCHUNK)       // deep prefetch into L2 (global_prefetch_b8)
                __builtin_prefetch((const char*)(src + (chunk + 2) * 1024) + tid * 64, 0, 1);
        }
#if HAVE_ASYNC_LDS
        // Async loads complete in order: <=4 outstanding means this chunk's 4 are done.
        if (chunk + 1 < NCHUNK) WAIT_ASYNC(4); else WAIT_ASYNC(0);
#endif
        __syncthreads();

        // ---- Two 16-wide v-tiles, interleaved as independent WMMA chains ----
        // B (4x16, K x N) operand: N = lane%16 selects doc row, half selects K pair.
        const float* brow0 = &ldsD[cur][( 0 + M) * LDS_STRIDE + kh * 2];
        const float* brow1 = &ldsD[cur][(16 + M) * LDS_STRIDE + kh * 2];
        v8f c0 = {}, c1 = {};
#pragma unroll
        for (int kk = 0; kk < 32; ++kk) {
            v2f b0 = *(const v2f*)(brow0 + kk * 4);
            v2f b1 = *(const v2f*)(brow1 + kk * 4);
            c0 = __builtin_amdgcn_wmma_f32_16x16x4_f32(
                    false, a[kk], false, b0, (short)0, c0, false, false);
            c1 = __builtin_amdgcn_wmma_f32_16x16x4_f32(
                    false, a[kk], false, b1, (short)0, c1, false, false);
        }
        // Elementwise running max: (u, lane-col) -> v mapping is fixed across tiles.
#pragma unroll
        for (int r = 0; r < 8; ++r)
            rowmax[r] = fmaxf(rowmax[r], fmaxf(c0[r], c1[r]));

        __syncthreads();   // buffer reused two iterations later
    }

    // ---- Reduce: max over v (across the 16 lanes of each half), then sum over u ----
    float s = 0.0f;
#pragma unroll
    for (int r = 0; r < 8; ++r) {
        float v = rowmax[r];
#pragma unroll
        for (int off = 1; off < 16; off <<= 1)
            v = fmaxf(v, __shfl_xor(v, off, 32));
        s += v;                      // half0 sums rows M=r, half1 rows M=r+8
    }
    s += __shfl_xor(s, 16, 32);      // combine halves -> sum over this wave's 16 u rows

    if (lane == 0) red[wave] = s;
    __syncthreads();
    if (tid == 0) {
        float t = 0.0f;
#pragma unroll
        for (int w = 0; w < WAVES; ++w) t += red[w];
        out[bid] = t;
    }
}

extern "C" void kernel_launch(void* const* d_in, const int* in_sizes, int n_in,
                              void* d_out, int out_size, void* d_ws, size_t ws_size,
                              hipStream_t stream) {
    const float* hq = (const float*)d_in[0];   // [16,4,128,128]
    const float* hd = (const float*)d_in[1];   // [16,4,16,256,128]
    float* out = (float*)d_out;                // [16,4,16] = 1024
    (void)in_sizes; (void)n_in; (void)d_ws; (void)ws_size;

    colbert_maxsim_kernel<<<dim3(out_size), dim3(256), 0, stream>>>(hq, hd, out);
}